// PadNonConstant_962072674565
// MI455X (gfx1250) — compile-verified
//
#include <hip/hip_runtime.h>

// Input : [16, 32, 512, 512] f32
// Output: [16, 46, 526, 518] f32  (7 fused pads, composed inverse index map)

#define IN_C  32
#define IN_H  512
#define IN_W  512
#define OUT_C 46
#define OUT_H 526
#define OUT_W 518
#define NBATCH 16

__global__ __launch_bounds__(256)
void pad_chain_fused_kernel(const float* __restrict__ x, float* __restrict__ out) {
    // One block per output row (n, c7, h7); threads stride over W.
    const int h7 = blockIdx.y;
    const int nc = blockIdx.z;
    const int n  = nc / OUT_C;          // constant-divide -> SALU mul/shift
    const int c7 = nc - n * OUT_C;

    // ---- channel inverse map (wave-uniform -> SALU) ----
    // step7: reflect (0,3) on 43 ; step6: edge (0,3) on 40
    // step5: wrap (3,3) on 34    ; step4: wrap (1,1) on 32
    int c6 = (c7 < 43) ? c7 : (84 - c7);
    int c5 = min(c6, 39);
    int c4 = (c5 + 31) % 34;
    int ci = (c4 + 31) & 31;

    // ---- height inverse map (wave-uniform -> SALU) ----
    // step7: reflect (3,0) on 523 ; step6: edge (3,0) on 520
    // step3: edge (2,2) on 516    ; step2: reflect (2,2) on 512
    int h6 = (h7 >= 3) ? (h7 - 3) : (3 - h7);
    int h5 = max(h6 - 3, 0);
    int h2 = min(max(h5 - 2, 0), 515);
    int ih = h2 - 2;
    int hi = (ih < 0) ? -ih : ((ih > 511) ? (1022 - ih) : ih);

    const float* __restrict__ src =
        x + (((size_t)n * IN_C + (size_t)ci) * IN_H + (size_t)hi) * IN_W;
    float* __restrict__ dst =
        out + (((size_t)n * OUT_C + (size_t)c7) * OUT_H + (size_t)h7) * OUT_W;

    // ---- width inverse map (per-lane VALU) + streaming copy ----
    // step7: reflect (1,1) on 516 ; step6: edge (1,1) on 514 ; step1: reflect (1,1) on 512
    for (int w7 = threadIdx.x; w7 < OUT_W; w7 += 256) {
        int i1 = w7 - 1;
        int w6 = (i1 < 0) ? -i1 : ((i1 > 515) ? (1030 - i1) : i1);
        int w5 = min(max(w6 - 1, 0), 513);
        int i0 = w5 - 1;
        int wi = (i0 < 0) ? -i0 : ((i0 > 511) ? (1022 - i0) : i0);

        float v = src[wi];                        // RT load: input slab has ~1.4x L2 reuse
        __builtin_nontemporal_store(v, dst + w7); // NT store: output is write-once, bypass L2 dirty residency
    }
}

extern "C" void kernel_launch(void* const* d_in, const int* in_sizes, int n_in,
                              void* d_out, int out_size, void* d_ws, size_t ws_size,
                              hipStream_t stream) {
    const float* x = (const float*)d_in[0];
    float* out = (float*)d_out;
    (void)in_sizes; (void)n_in; (void)out_size; (void)d_ws; (void)ws_size;

    // grid: (1, H_out, N*C_out) -> 16*46*526 = 387,136 blocks, 256 threads each
    dim3 grid(1, OUT_H, NBATCH * OUT_C);
    pad_chain_fused_kernel<<<grid, dim3(256, 1, 1), 0, stream>>>(x, out);
}